// CIF_6399501271103
// MI455X (gfx1250) — compile-verified
//
#include <hip/hip_runtime.h>
#include <stdint.h>

// ---------------- problem constants ----------------
#define B_    32
#define T_    2000
#define H_    512
#define LOUT  250
#define THRF  0.95f

// ---------------- tiling ----------------
#define LG    16            // output slots per l-group (WMMA M)
#define NLG   16            // 16*16 = 256 >= 250 slots
#define HB    256           // h columns per block
#define NHB   (H_/HB)       // 2
#define KT    128           // staged t-rows per LDS chunk
#define LDSS  (HB+4)        // padded LDS row stride in floats (kills bank conflicts)
#define FCAP  256           // fire-record capacity per batch row (>= 250)

typedef __attribute__((ext_vector_type(2))) float v2f;
typedef __attribute__((ext_vector_type(8))) float v8f;
typedef __attribute__((ext_vector_type(4))) float f4;
typedef __attribute__((ext_vector_type(4))) int   v4i;

#if defined(__has_builtin)
#  if __has_builtin(__builtin_amdgcn_global_load_async_to_lds_b128)
#    define HAVE_ASYNC 1
#  endif
#  if __has_builtin(__builtin_amdgcn_wmma_f32_16x16x4_f32)
#    define HAVE_WMMA 1
#  endif
#  if __has_builtin(__builtin_amdgcn_s_wait_asynccnt)
#    define HAVE_WAITASYNC 1
#  endif
#endif
#ifndef HAVE_ASYNC
#define HAVE_ASYNC 0
#endif
#ifndef HAVE_WMMA
#define HAVE_WMMA 0
#endif
#ifndef HAVE_WAITASYNC
#define HAVE_WAITASYNC 0
#endif

// =====================================================================
// Pass 1: exact sequential CIF recurrence per batch row.
// One wave32; lane b scans row b with the reference's f32 op order, so
// fire decisions match the reference bit-for-bit.
// =====================================================================
__global__ __launch_bounds__(32) void cif_scan(
    const float* __restrict__ alphas,
    int*   __restrict__ nf,       // [B]
    int*   __restrict__ fireT,    // [B][FCAP] fire time t
    float* __restrict__ fireC,    // [B][FCAP] cur   = 1 - integrate_before
    float* __restrict__ fireR)    // [B][FCAP] remainds = alpha - cur
{
    int b = threadIdx.x;
    if (b >= B_) return;
    const float* a = alphas + b * T_;
    float integ = 0.0f;
    int   n = 0;
    for (int t = 0; t < T_; ++t) {
        float al = a[t];
        float i2 = integ + al;          // integ = integrate + alpha
        bool fire = i2 > THRF;
        if (fire) {
            float cur = 1.0f - integ;   // dist_comp
            if (n < FCAP) {
                fireT[b * FCAP + n] = t;
                fireC[b * FCAP + n] = cur;
                fireR[b * FCAP + n] = al - cur;
            }
            ++n;
            integ = i2 - 1.0f;
        } else {
            integ = i2;
        }
    }
    nf[b] = n;
}

// band weight of slot (with params sM,eM,cM,rM) at time t
__device__ __forceinline__ float wgt(int t, int sM, int eM, float cM, float rM,
                                     const float* __restrict__ aT, int k0)
{
    float al = aT[t - k0];
    float w = (t > sM && t < eM) ? al : 0.0f;  // interior: raw alpha
    w = (t == eM) ? cM : w;                    // own fire step: cur
    w = (t == sM) ? rM : w;                    // previous fire step: remainds carry
    return w;
}

// =====================================================================
// Pass 2: out[b, l0:l0+16, hbase:hbase+256] = W_band @ hidden via WMMA.
// hidden rows streamed through LDS with CDNA5 async-to-LDS copies.
// =====================================================================
__global__ __launch_bounds__(128) void cif_emit(
    const float* __restrict__ hidden,
    const float* __restrict__ alphas,
    float* __restrict__ out,
    const int*   __restrict__ nf,
    const int*   __restrict__ fireT,
    const float* __restrict__ fireC,
    const float* __restrict__ fireR)
{
    extern __shared__ float smem[];
    float* tile = smem;                 // [KT][LDSS] staged hidden rows
    float* aT   = tile + KT * LDSS;     // [KT]      staged alphas
    int*   sE   = (int*)(aT + KT);      // [LG] fire t of slot
    int*   sS   = sE + LG;              // [LG] fire t of previous slot (-1 for l==0)
    float* sC   = (float*)(sS + LG);    // [LG] cur at own fire
    float* sR   = sC + LG;              // [LG] remainds carry from previous fire

    const int hb = blockIdx.x;
    const int lg = blockIdx.y;
    const int b  = blockIdx.z;
    const int l0 = lg * LG;
    const int hbase = hb * HB;
    const int tid  = threadIdx.x;
    const int lane = tid & 31;
    const int wv   = tid >> 5;           // wave id 0..3
    const int n    = lane & 15;          // N column within tile / M row for A
    const int kB   = (lane < 16) ? 0 : 2;// A/B K pair per ISA VGPR layout

    const int F  = nf[b];
    const int Fc = (F < LOUT) ? F : LOUT;

    if (tid < LG) {
        int l = l0 + tid;
        bool sv = (l < Fc);
        sE[tid] = sv ? fireT[b * FCAP + l] : -5;
        sS[tid] = sv ? ((l == 0) ? -1 : fireT[b * FCAP + l - 1]) : -5;
        sC[tid] = sv ? fireC[b * FCAP + l] : 0.0f;
        sR[tid] = (sv && l > 0) ? fireR[b * FCAP + l - 1] : 0.0f;
    }
    __syncthreads();

    const int   eM = sE[n];
    const int   sM = sS[n];
    const float cM = sC[n];
    const float rM = sR[n];

    // block's t-range: rows feeding any valid slot in this group
    int lmax = l0 + LG - 1;
    if (lmax > Fc - 1) lmax = Fc - 1;
    int tmin = 0, tmax = -1;
    if (lmax >= l0) {
        tmin = (l0 == 0) ? 0 : fireT[b * FCAP + l0 - 1];
        tmax = fireT[b * FCAP + lmax];
    }

    v8f acc[4] = {};   // h-tiles wv, wv+4, wv+8, wv+12

    for (int k0 = tmin; k0 <= tmax; k0 += KT) {
        int rows = tmax - k0 + 1; if (rows > KT) rows = KT;
        int rows4 = (rows + 3) & ~3;
        // zero-fill K tail rows: WMMA propagates NaN, 0 * garbage is unsafe
        for (int r = rows; r < rows4; ++r) {
            tile[r * LDSS + 2 * tid]     = 0.0f;
            tile[r * LDSS + 2 * tid + 1] = 0.0f;
        }
        if (tid < rows) aT[tid] = alphas[b * T_ + k0 + tid];

        // stage hidden[b, k0:k0+rows, hbase:hbase+HB] -> LDS (async b128 per lane)
        const float* src0 = hidden + ((size_t)(b * T_ + k0)) * H_ + hbase;
        for (int g = tid; g < rows * (HB / 4); g += 128) {
            int r  = g >> 6;             // HB/4 = 64 float4 groups per row
            int c4 = (g & 63) << 2;
            const float* src = src0 + (size_t)r * H_ + c4;
            float* dst = tile + r * LDSS + c4;
#if HAVE_ASYNC
            __builtin_amdgcn_global_load_async_to_lds_b128(
                (__attribute__((address_space(1))) v4i*)src,
                (__attribute__((address_space(3))) v4i*)dst, 0, 0);
#else
            *(f4*)dst = *(const f4*)src;
#endif
        }
#if HAVE_ASYNC
#  if HAVE_WAITASYNC
        __builtin_amdgcn_s_wait_asynccnt(0);
#  else
        asm volatile("s_wait_asynccnt 0" ::: "memory");
#  endif
#endif
        __syncthreads();

        for (int kc = 0; kc < rows4; kc += 4) {
            // A (16x4 f32) per ISA layout: lanes 0-15 hold K={0,1}, 16-31 K={2,3}; M = lane&15
            int tA = k0 + kc + kB;
            v2f A;
            A.x = wgt(tA,     sM, eM, cM, rM, aT, k0);
            A.y = wgt(tA + 1, sM, eM, cM, rM, aT, k0);
            const float* brow = tile + (kc + kB) * LDSS + n;
#pragma unroll
            for (int i = 0; i < 4; ++i) {
                int cb = (wv + 4 * i) * 16;
                // B (4x16 f32): VGPR0 = K rows {0,2}, VGPR1 = K rows {1,3}, N = lane&15
                v2f Bv;
                Bv.x = brow[cb];
                Bv.y = brow[LDSS + cb];
#if HAVE_WMMA
                acc[i] = __builtin_amdgcn_wmma_f32_16x16x4_f32(
                    false, A, false, Bv, (short)0, acc[i], false, false);
#else
                int mr = (lane < 16) ? 0 : 8;
#pragma unroll
                for (int j = 0; j < 8; ++j) {
                    int mm = j + mr;
                    float s = 0.0f;
#pragma unroll
                    for (int k = 0; k < 4; ++k) {
                        float w = wgt(k0 + kc + k, sS[mm], sE[mm], sC[mm], sR[mm], aT, k0);
                        s += w * tile[(kc + k) * LDSS + cb + n];
                    }
                    acc[i][j] += s;
                }
#endif
            }
        }
        __syncthreads();  // before restaging overwrites the tile
    }

    // store D: VGPR j -> row M=j (lanes 0-15) / j+8 (lanes 16-31), col N=lane&15
    const int mrow = (lane < 16) ? 0 : 8;
#pragma unroll
    for (int i = 0; i < 4; ++i) {
        int h = hbase + (wv + 4 * i) * 16 + n;
#pragma unroll
        for (int j = 0; j < 8; ++j) {
            int lo = l0 + j + mrow;
            if (lo < LOUT)
                out[((size_t)b * LOUT + lo) * H_ + h] = acc[i][j];
        }
    }
}

extern "C" void kernel_launch(void* const* d_in, const int* in_sizes, int n_in,
                              void* d_out, int out_size, void* d_ws, size_t ws_size,
                              hipStream_t stream)
{
    const float* hidden = (const float*)d_in[0];   // [B,T,H] f32
    const float* alphas = (const float*)d_in[1];   // [B,T]   f32
    float* out = (float*)d_out;                    // [B,LOUT,H] f32

    // workspace layout (~100 KB)
    int*   nf    = (int*)d_ws;                         // [B] (padded to 64)
    int*   fireT = nf + 64;                            // [B][FCAP]
    float* fireC = (float*)(fireT + B_ * FCAP);        // [B][FCAP]
    float* fireR = fireC + B_ * FCAP;                  // [B][FCAP]

    cif_scan<<<1, 32, 0, stream>>>(alphas, nf, fireT, fireC, fireR);

    dim3 grid(NHB, NLG, B_);
    size_t lds = (size_t)(KT * LDSS + KT) * sizeof(float)
               + (size_t)LG * (2 * sizeof(int) + 2 * sizeof(float));
    cif_emit<<<grid, 128, lds, stream>>>(hidden, alphas, out, nf, fireT, fireC, fireR);
}